// DifferentialQuadraticSplineStack_71657234366526
// MI455X (gfx1250) — compile-verified
//
#include <hip/hip_runtime.h>
#include <hip/hip_bf16.h>
#include <stdint.h>

// ---------------------------------------------------------------------------
// DifferentialQuadraticSplineStack for MI455X (gfx1250, wave32)
//   NBINS = (128, 64, 32), SUM_H = 224, SUM_W = 221
//   Phase 1: per-gene softmax widths + bin locations + gathered height logits
//   Phase 2: one wave32 per point; lanes own contiguous bin chunks;
//            wave-scan cumsum, wave-uniform bin search, shuffle gathers.
//   delta rows (896B) double-buffered through LDS via CDNA5 async copies:
//   issue row p+stride, s_wait_asynccnt 2 (in-order completion), consume row p.
// ---------------------------------------------------------------------------

static constexpr int NB0 = 128, NB1 = 64, NB2 = 32;
static constexpr int SUM_H = 224;  // 128+64+32
static constexpr int SUM_W = 221;  // 127+63+31

#define WPB 8            // waves per block
#define TPB (WPB * 32)   // 256 threads

#if defined(__HIP_DEVICE_COMPILE__) && defined(__has_builtin)
#if __has_builtin(__builtin_amdgcn_global_load_async_to_lds_b128)
#define HAVE_ASYNC 1
#endif
#endif
#ifndef HAVE_ASYNC
#define HAVE_ASYNC 0
#endif

// b128 async copy moves one int4 per active lane; params are
// (v4i addrspace(1)* gsrc, v4i addrspace(3)* ldst, imm offset, imm cpol)
typedef int v4i __attribute__((vector_size(4 * sizeof(int))));
typedef __attribute__((address_space(1))) v4i gas_v4i;  // global
typedef __attribute__((address_space(3))) v4i las_v4i;  // LDS

template <int CNT>
__device__ __forceinline__ void wait_asynccnt_le() {
#if defined(__HIP_DEVICE_COMPILE__)
#if defined(__has_builtin) && __has_builtin(__builtin_amdgcn_s_wait_asynccnt)
  __builtin_amdgcn_s_wait_asynccnt(CNT);
#else
  asm volatile("s_wait_asynccnt %0" ::"i"(CNT) : "memory");
#endif
#endif
  asm volatile("" ::: "memory");
}

#if HAVE_ASYNC
// Stage one 224-float (896B) delta row into LDS: 2 ASYNCcnt increments.
__device__ __forceinline__ void issue_row_copy(const float* src, float* dst,
                                               int lane) {
  __builtin_amdgcn_global_load_async_to_lds_b128(
      (gas_v4i*)(uintptr_t)(src + lane * 4),
      (las_v4i*)(uintptr_t)(dst + lane * 4), 0, 0);
  if (lane < 24) {
    __builtin_amdgcn_global_load_async_to_lds_b128(
        (gas_v4i*)(uintptr_t)(src + 128 + lane * 4),
        (las_v4i*)(uintptr_t)(dst + 128 + lane * 4), 0, 0);
  }
}
#endif

// ---------------------------------------------------------------------------
// Phase 1: one thread per gene-of-interest. Computes
//   uh_g[g][0:224]  = heights_weight[genes_oi[g]]
//   w_g [g][0:221]  = softmax(widths_weight[genes_oi[g]]) per level slice
//   bl_g[g][0:224]  = padded cumsum of widths, last of each level forced 1.0
// ---------------------------------------------------------------------------
__global__ void __launch_bounds__(256) precompute_genes(
    const float* __restrict__ heights_weight,
    const float* __restrict__ widths_weight,
    const int*   __restrict__ genes_oi,
    float* __restrict__ uh_g,
    float* __restrict__ w_g,
    float* __restrict__ bl_g,
    int ngoi)
{
  int g = blockIdx.x * blockDim.x + threadIdx.x;
  if (g >= ngoi) return;
  int gene = genes_oi[g];
  const float* hw = heights_weight + (size_t)gene * SUM_H;
  const float* ww = widths_weight  + (size_t)gene * SUM_W;
  float* uh = uh_g + (size_t)g * SUM_H;
  float* wg = w_g  + (size_t)g * SUM_W;
  float* bl = bl_g + (size_t)g * SUM_H;

  for (int t = 0; t < SUM_H; ++t) uh[t] = hw[t];

  const int nbins[3] = {NB0, NB1, NB2};
  int w_off = 0, h_off = 0;
  for (int l = 0; l < 3; ++l) {
    const int n = nbins[l], m = n - 1;
    float mx = -3.4e38f;
    for (int i = 0; i < m; ++i) mx = fmaxf(mx, ww[w_off + i]);
    float s = 0.0f;
    for (int i = 0; i < m; ++i) {
      float e = expf(ww[w_off + i] - mx);
      wg[w_off + i] = e;
      s += e;
    }
    float inv = 1.0f / s;
    float c = 0.0f;
    bl[h_off] = 0.0f;
    for (int i = 0; i < m; ++i) {
      float wv = wg[w_off + i] * inv;
      wg[w_off + i] = wv;
      c += wv;
      bl[h_off + i + 1] = (i == m - 1) ? 1.0f : c;
    }
    w_off += m;
    h_off += n;
  }
}

// ---------------------------------------------------------------------------
// Wave-uniform gather: value at flat bin-axis index `idx`, where lane L holds
// elements [L*VPL, L*VPL+VPL). idx is uniform across the wave.
// ---------------------------------------------------------------------------
template <int VPL>
__device__ __forceinline__ float wave_gather(const float (&v)[VPL], int idx) {
  const int src = idx / VPL;
  const int sub = idx % VPL;
  float r = __shfl(v[0], src, 32);
#pragma unroll
  for (int j = 1; j < VPL; ++j) {
    float t = __shfl(v[j], src, 32);
    r = (sub == j) ? t : r;
  }
  return r;
}

// ---------------------------------------------------------------------------
// One spline level for one point, processed by one wave32.
// ---------------------------------------------------------------------------
template <int N>
__device__ __forceinline__ void spline_level(
    int lane, int g, int h_off, int w_off,
    const float* __restrict__ uh_g,
    const float* __restrict__ w_g,
    const float* __restrict__ bl_g,
    const float* drow,            // delta row (LDS or global)
    float& x, float& lad)
{
  constexpr int VPL = N / 32;
  const int k0 = lane * VPL;

  const float* uhp = uh_g + (size_t)g * SUM_H + h_off + k0;
  const float* dd  = drow + h_off + k0;
  const float* wp  = w_g  + (size_t)g * SUM_W + w_off + k0;
  const float* blp = bl_g + (size_t)g * SUM_H + h_off + k0;

  float e[VPL], wl[VPL], blv[VPL];
#pragma unroll
  for (int j = 0; j < VPL; ++j) {
    e[j]   = expf(uhp[j] + dd[j]);
    blv[j] = blp[j];
    wl[j]  = 0.0f;
    if (k0 + j < N - 1) wl[j] = wp[j];  // width for bin k (k <= N-2)
  }

  // trapezoid terms t[k] = 0.5*(e[k]+e[k+1])*w[k]; t[N-1] = 0 via wl = 0
  float e_next_chunk = __shfl_down(e[0], 1, 32);  // e[(lane+1)*VPL]
  float t[VPL];
#pragma unroll
  for (int j = 0; j < VPL; ++j) {
    float en = (j < VPL - 1) ? e[(j + 1 < VPL) ? j + 1 : 0] : e_next_chunk;
    t[j] = 0.5f * (e[j] + en) * wl[j];
  }

  // local inclusive prefix within chunk + wave inclusive scan of chunk sums
  float pre[VPL];
  float run = 0.0f;
#pragma unroll
  for (int j = 0; j < VPL; ++j) { run += t[j]; pre[j] = run; }
  float v = run;
#pragma unroll
  for (int d = 1; d < 32; d <<= 1) {
    float u = __shfl_up(v, d, 32);
    if (lane >= d) v += u;
  }
  const float excl = v - run;             // sum of t over all k < k0
  const float area = __shfl(v, 31, 32);   // total trapezoid area
  const float invA = 1.0f / area;

  // left CDF of each bin owned by this lane (bin N-1's forced 1.0 never read)
  float cdfv[VPL];
#pragma unroll
  for (int j = 0; j < VPL; ++j)
    cdfv[j] = (excl + (j > 0 ? pre[j - 1] : 0.0f)) * invA;

  // bin search: count of bin_locations <= x (x is wave-uniform)
  int cnt = 0;
#pragma unroll
  for (int j = 0; j < VPL; ++j) cnt += (x >= blv[j]) ? 1 : 0;
#pragma unroll
  for (int d = 16; d > 0; d >>= 1) cnt += __shfl_xor(cnt, d, 32);
  int bin = cnt - 1;
  bin = bin < 0 ? 0 : (bin > N - 2 ? N - 2 : bin);

  // wave-uniform gathers
  const float in_bl  = wave_gather<VPL>(blv,  bin);
  const float in_cdf = wave_gather<VPL>(cdfv, bin);
  const float wv     = wave_gather<VPL>(wl,   bin);
  const float hl     = wave_gather<VPL>(e,    bin)     * invA;
  const float hr     = wave_gather<VPL>(e,    bin + 1) * invA;

  const float alpha = (x - in_bl) / wv;
  const float aa = 0.5f * (hr - hl) * wv;
  const float bb = hl * wv;
  float out = (aa * alpha + bb) * alpha + in_cdf;
  out = fminf(fmaxf(out, 0.0f), 1.0f);
  x = out;
  lad += logf(alpha * (hr - hl) + hl);
}

// ---------------------------------------------------------------------------
// Phase 2: grid-stride over points; one wave per point; double-buffered
// async global->LDS staging of the delta rows.
// ---------------------------------------------------------------------------
__global__ void __launch_bounds__(TPB) spline_forward(
    const float* __restrict__ x_in,
    const float* __restrict__ delta,
    const int*   __restrict__ lgi,
    const float* __restrict__ uh_g,
    const float* __restrict__ w_g,
    const float* __restrict__ bl_g,
    float* __restrict__ out,
    float* __restrict__ lad_out,
    int npoints)
{
#if HAVE_ASYNC
  __shared__ __align__(16) float sdelta[WPB][2][SUM_H];  // 2 x 896 B per wave
#endif
  const int lane    = threadIdx.x & 31;
  const int wslot   = threadIdx.x >> 5;
  const int wave_id = blockIdx.x * WPB + wslot;
  const int stride  = gridDim.x * WPB;

#if HAVE_ASYNC
  float (&buf)[2][SUM_H] = sdelta[wslot];
  int parity = 0;
  int p = wave_id;
  if (p < npoints) issue_row_copy(delta + (size_t)p * SUM_H, buf[0], lane);
  for (; p < npoints; p += stride) {
    const int pn = p + stride;
    if (pn < npoints) {
      // issue next row into the other buffer, then wait only for the
      // current row (async loads complete in order -> cnt<=2 suffices)
      issue_row_copy(delta + (size_t)pn * SUM_H, buf[parity ^ 1], lane);
      if (pn + stride < npoints)  // warm L2 two rows ahead
        __builtin_prefetch(delta + (size_t)(pn + stride) * SUM_H + lane * 7, 0, 1);
      wait_asynccnt_le<2>();
    } else {
      wait_asynccnt_le<0>();
    }

    const int g = lgi[p];
    float x   = x_in[p];
    float lad = 0.0f;
    const float* drow = &buf[parity][0];

    spline_level<NB0>(lane, g,   0,   0, uh_g, w_g, bl_g, drow, x, lad);
    spline_level<NB1>(lane, g, 128, 127, uh_g, w_g, bl_g, drow, x, lad);
    spline_level<NB2>(lane, g, 192, 190, uh_g, w_g, bl_g, drow, x, lad);

    if (lane == 0) {
      out[p]     = x;
      lad_out[p] = lad;
    }
    parity ^= 1;
  }
#else
  for (int p = wave_id; p < npoints; p += stride) {
    const float* drow = delta + (size_t)p * SUM_H;
    if (p + stride < npoints)
      __builtin_prefetch(delta + (size_t)(p + stride) * SUM_H + lane * 7, 0, 1);

    const int g = lgi[p];
    float x   = x_in[p];
    float lad = 0.0f;

    spline_level<NB0>(lane, g,   0,   0, uh_g, w_g, bl_g, drow, x, lad);
    spline_level<NB1>(lane, g, 128, 127, uh_g, w_g, bl_g, drow, x, lad);
    spline_level<NB2>(lane, g, 192, 190, uh_g, w_g, bl_g, drow, x, lad);

    if (lane == 0) {
      out[p]     = x;
      lad_out[p] = lad;
    }
  }
#endif
}

// ---------------------------------------------------------------------------
extern "C" void kernel_launch(void* const* d_in, const int* in_sizes, int n_in,
                              void* d_out, int out_size, void* d_ws, size_t ws_size,
                              hipStream_t stream) {
  const float* x        = (const float*)d_in[0];
  const float* delta    = (const float*)d_in[1];
  const float* hweight  = (const float*)d_in[2];
  const float* wweight  = (const float*)d_in[3];
  const int*   genes_oi = (const int*)d_in[4];
  const int*   lgi      = (const int*)d_in[5];
  const int npoints = in_sizes[0];
  const int ngoi    = in_sizes[4];

  // workspace layout (floats): uh_g[ngoi*224] | w_g[ngoi*221] | bl_g[ngoi*224]
  float* uh_g = (float*)d_ws;
  float* w_g  = uh_g + (size_t)ngoi * SUM_H;
  float* bl_g = w_g  + (size_t)ngoi * SUM_W;

  float* out = (float*)d_out;
  float* lad = out + npoints;

  precompute_genes<<<(ngoi + 255) / 256, 256, 0, stream>>>(
      hweight, wweight, genes_oi, uh_g, w_g, bl_g, ngoi);

  int blocks = (npoints + WPB - 1) / WPB;
  if (blocks > 4096) blocks = 4096;
  spline_forward<<<blocks, TPB, 0, stream>>>(
      x, delta, lgi, uh_g, w_g, bl_g, out, lad, npoints);
}